// AdaptiveFrequencyDecomposition_24567212933355
// MI455X (gfx1250) — compile-verified
//
#include <hip/hip_runtime.h>
#include <math.h>

typedef __attribute__((ext_vector_type(2))) float v2f;
typedef __attribute__((ext_vector_type(4))) float v4f;
typedef __attribute__((ext_vector_type(8))) float v8f;

#define LROW 20   // padded LDS row (floats): 16B-aligned, avoids bank conflicts

__device__ __forceinline__ float dct8(int k, int i) {
    // D[k][i] for 8-point DCT-II (orthonormal)
    if (k == 0) return 0.35355339059327373f;               // 1/sqrt(8)
    return 0.5f * __cosf(0.19634954084936207f * (float)(k * (2 * i + 1))); // sqrt(2/8)*cos(pi*k*(2i+1)/16)
}
__device__ __forceinline__ float Emat(int r, int c) {
    // E = blockdiag(D, D), 16x16
    float v = 0.0f;
    if (r < 8 && c < 8)        v = dct8(r, c);
    else if (r >= 8 && c >= 8) v = dct8(r - 8, c - 8);
    return v;
}
__device__ __forceinline__ float zigzag(int zr, int zc) {
    // normalized zigzag index, matches reference generator
    int d = zr + zc;
    int idx;
    if (d < 8) {
        int base = d * (d + 1) / 2;
        idx = base + ((d & 1) ? zr : zc);
    } else {
        int base = 64 - (15 - d) * (16 - d) / 2;
        idx = base + ((d & 1) ? (zr - (d - 7)) : (7 - zr));
    }
    return (float)idx * (1.0f / 63.0f);
}
__device__ __forceinline__ float sigmoidf(float x) { return 1.0f / (1.0f + __expf(-x)); }

// ---------------- Stage 1a: AdaptiveAvgPool2d(8): one block per pooled cell ----
__global__ void afd_pool_kernel(const float* __restrict__ x, float* __restrict__ pooled) {
    __shared__ float sm[256];
    int bid = blockIdx.x;                       // b*192 + c*64 + i*8 + j
    int j = bid & 7, i = (bid >> 3) & 7;
    int bc = bid >> 6;                          // b*3 + c
    int c = bc % 3, b = bid / 192;
    const float* base = x + (((size_t)(b * 3 + c) * 512 + (size_t)i * 64) * 512 + (size_t)j * 64);
    int t = threadIdx.x;
    float s = 0.f;
    #pragma unroll
    for (int k = 0; k < 16; ++k) {
        int e = t + 256 * k;                    // coalesced over 64x64 region
        s += base[(e >> 6) * 512 + (e & 63)];
    }
    sm[t] = s;
    __syncthreads();
    for (int w = 128; w > 0; w >>= 1) { if (t < w) sm[t] += sm[t + w]; __syncthreads(); }
    if (t == 0) pooled[bid] = sm[0] * (1.0f / 4096.0f);
}

// ---------------- Stage 1b: tiny MLP -> per-batch thresholds -------------------
__global__ void afd_mlp_kernel(const float* __restrict__ pooled,
                               const float* __restrict__ w1, const float* __restrict__ b1,
                               const float* __restrict__ w2, const float* __restrict__ b2,
                               float* __restrict__ thr /* [32]: low_t[16], high_t[16] */) {
    __shared__ float h[16][32];
    int t = threadIdx.x;
    int b = t >> 5, n = t & 31;                 // 512 threads = 16 batches x 32 hidden
    float s = b1[n];
    for (int k = 0; k < 192; ++k) s += pooled[b * 192 + k] * w1[k * 32 + n];
    h[b][n] = fmaxf(s, 0.0f);
    __syncthreads();
    if (n < 2) {
        float s2 = b2[n];
        #pragma unroll
        for (int k = 0; k < 32; ++k) s2 += h[b][k] * w2[k * 2 + n];
        float sg = sigmoidf(s2);
        if (n == 0) thr[b]      = sg * 0.25f + 0.15f;   // low_t in [0.15,0.4]
        else        thr[16 + b] = sg * 0.25f + 0.60f;   // high_t in [0.6,0.85]
    }
}

// ---------------- Stage 2: DCT / mask / IDCT via V_WMMA_F32_16X16X4_F32 --------
// One wave = one 16x16 pixel tile (= 2x2 DCT blocks) via E = blockdiag(D,D):
//   Y = E X E^T ; Z_m = E^T (Y .* M_m) E  for m in {low, mid, high}
__global__ void __launch_bounds__(256) afd_freq_kernel(
    const float* __restrict__ x, const float* __restrict__ thr,
    float* __restrict__ out_low, float* __restrict__ out_mid, float* __restrict__ out_high)
{
    __shared__ float ldsAll[8][16 * LROW];      // 10 KB/block, one slice per wave
    const int lane = threadIdx.x & 31;
    const int wv   = threadIdx.x >> 5;
    float* L = ldsAll[wv];

    int tile = blockIdx.x * 8 + wv;             // 49152 tiles total
    int tx = tile & 31;
    int ty = (tile >> 5) & 31;
    int c  = (tile >> 10) % 3;
    int b  = tile / 3072;

    const int m  = lane & 15;                   // row (A) / col (B,C,D) index
    const int hi = lane >> 4;
    const int n  = m;

    // Constant WMMA operands for E:
    //  P[s] : E as A-operand for (E @ X); also E^T as B-operand (since n==m)
    //  Q[s] : E^T as A-operand for (E^T @ Ym); also E as B-operand
    v2f P[4], Q[4];
    #pragma unroll
    for (int s = 0; s < 4; ++s) {
        int k0 = 4 * s + 2 * hi;
        v2f p; p.x = Emat(m, k0);     p.y = Emat(m, k0 + 1);     P[s] = p;
        v2f q; q.x = Emat(k0, m);     q.y = Emat(k0 + 1, m);     Q[s] = q;
    }

    float lt = thr[b], ht = thr[16 + b];
    float lowm[8], highm[8];
    #pragma unroll
    for (int r = 0; r < 8; ++r) {               // actual row = r + 8*hi; (row%8)==r
        float zz = zigzag(r, n & 7);
        lowm[r]  = sigmoidf(50.0f * (lt - zz));
        highm[r] = sigmoidf(50.0f * (zz - ht));
    }

    size_t base = ((size_t)(b * 3 + c) * 512 + (size_t)ty * 16) * 512 + (size_t)tx * 16;

    // Load 16x16 tile -> LDS (row-major, padded), fully coalesced b128 loads
    #pragma unroll
    for (int k = 0; k < 2; ++k) {
        int q  = k * 32 + lane;                 // 64 float4s
        int row = q >> 2, c4 = q & 3;
        v4f v = *(const v4f*)(x + base + (size_t)row * 512 + c4 * 4);
        *(v4f*)&L[row * LROW + c4 * 4] = v;
    }
    asm volatile("s_wait_dscnt 0x0" ::: "memory");

    // T = E @ X   (A = P const, B = X from LDS)
    v8f T = {};
    #pragma unroll
    for (int s = 0; s < 4; ++s) {
        int k0 = 4 * s + 2 * hi;
        v2f bx; bx.x = L[k0 * LROW + n]; bx.y = L[(k0 + 1) * LROW + n];
        T = __builtin_amdgcn_wmma_f32_16x16x4_f32(false, P[s], false, bx, (short)0, T, false, false);
    }
    // transpose T via LDS, then Y = T @ E^T  (A = T, B = P)
    #pragma unroll
    for (int r = 0; r < 8; ++r) L[(r + 8 * hi) * LROW + n] = T[r];
    asm volatile("s_wait_dscnt 0x0" ::: "memory");
    v8f Y = {};
    #pragma unroll
    for (int s = 0; s < 4; ++s) {
        int k0 = 4 * s + 2 * hi;
        v2f at; at.x = L[m * LROW + k0]; at.y = L[m * LROW + k0 + 1];
        Y = __builtin_amdgcn_wmma_f32_16x16x4_f32(false, at, false, P[s], (short)0, Y, false, false);
    }

    float* const outs[3] = {out_low, out_mid, out_high};
    #pragma unroll
    for (int mi = 0; mi < 3; ++mi) {
        // masked coefficients -> LDS (B-layout source)
        #pragma unroll
        for (int r = 0; r < 8; ++r) {
            float mk = (mi == 0) ? lowm[r]
                     : (mi == 2) ? highm[r]
                     : fmaxf(1.0f - lowm[r] - highm[r], 0.0f);
            L[(r + 8 * hi) * LROW + n] = Y[r] * mk;
        }
        asm volatile("s_wait_dscnt 0x0" ::: "memory");
        // U = E^T @ Ym   (A = Q const, B = Ym from LDS)
        v8f U = {};
        #pragma unroll
        for (int s = 0; s < 4; ++s) {
            int k0 = 4 * s + 2 * hi;
            v2f by; by.x = L[k0 * LROW + n]; by.y = L[(k0 + 1) * LROW + n];
            U = __builtin_amdgcn_wmma_f32_16x16x4_f32(false, Q[s], false, by, (short)0, U, false, false);
        }
        // transpose U via LDS, then Z = U @ E  (A = U, B = Q)
        #pragma unroll
        for (int r = 0; r < 8; ++r) L[(r + 8 * hi) * LROW + n] = U[r];
        asm volatile("s_wait_dscnt 0x0" ::: "memory");
        v8f Z = {};
        #pragma unroll
        for (int s = 0; s < 4; ++s) {
            int k0 = 4 * s + 2 * hi;
            v2f au; au.x = L[m * LROW + k0]; au.y = L[m * LROW + k0 + 1];
            Z = __builtin_amdgcn_wmma_f32_16x16x4_f32(false, au, false, Q[s], (short)0, Z, false, false);
        }
        float* o = outs[mi];
        #pragma unroll
        for (int r = 0; r < 8; ++r)
            o[base + (size_t)(r + 8 * hi) * 512 + n] = Z[r];
        asm volatile("s_wait_dscnt 0x0" ::: "memory"); // LDS reuse next iter
    }
}

extern "C" void kernel_launch(void* const* d_in, const int* in_sizes, int n_in,
                              void* d_out, int out_size, void* d_ws, size_t ws_size,
                              hipStream_t stream) {
    const float* x  = (const float*)d_in[0];
    const float* w1 = (const float*)d_in[1];
    const float* b1 = (const float*)d_in[2];
    const float* w2 = (const float*)d_in[3];
    const float* b2 = (const float*)d_in[4];
    float* out = (float*)d_out;
    float* pooled = (float*)d_ws;               // 3072 floats of scratch

    const size_t IMG = (size_t)16 * 3 * 512 * 512;
    float* out_low  = out;
    float* out_mid  = out + IMG;
    float* out_high = out + 2 * IMG;
    float* thr      = out + 3 * IMG;            // low_t[16], high_t[16] (output tail)

    afd_pool_kernel<<<3072, 256, 0, stream>>>(x, pooled);
    afd_mlp_kernel<<<1, 512, 0, stream>>>(pooled, w1, b1, w2, b2, thr);
    afd_freq_kernel<<<6144, 256, 0, stream>>>(x, thr, out_low, out_mid, out_high);
}